// RoiPoolingConv_55697135895302
// MI455X (gfx1250) — compile-verified
//
#include <hip/hip_runtime.h>
#include <cstdint>

#define POOL_P   7
#define NUM_ROIS 512
#define IMG_H    128
#define IMG_W    128
#define CHAN     1024
#define THREADS  256   // 256 threads * float4 = 1024 channels, one pass

typedef int   v4i_t __attribute__((ext_vector_type(4)));
typedef float v4f_t __attribute__((ext_vector_type(4)));

// ---------------------------------------------------------------------------
// CDNA5 async global->LDS copy (ASYNCcnt-tracked DMA), 16 bytes per lane.
// Builtin signature (probe-confirmed): (int4 AS1*, int4 AS3*, imm off, imm cpol)
// ---------------------------------------------------------------------------
__device__ __forceinline__ void async_copy_b128(const float* gsrc, float* ldst) {
#if __has_builtin(__builtin_amdgcn_global_load_async_to_lds_b128)
  typedef __attribute__((address_space(1))) v4i_t* gp_t;
  typedef __attribute__((address_space(3))) v4i_t* lp_t;
  __builtin_amdgcn_global_load_async_to_lds_b128(
      (gp_t)(uintptr_t)gsrc,
      (lp_t)(uint32_t)(uintptr_t)ldst,
      0, 0);
#else
  unsigned lds_off = (unsigned)(uintptr_t)ldst;   // low 32 bits = LDS byte offset
  asm volatile("global_load_async_to_lds_b128 %0, %1, off"
               :
               : "v"(lds_off), "v"(gsrc)
               : "memory");
#endif
}

template <int N>
__device__ __forceinline__ void wait_asynccnt() {
#if __has_builtin(__builtin_amdgcn_s_wait_asynccnt)
  __builtin_amdgcn_s_wait_asynccnt(N);
  asm volatile("" ::: "memory");
#else
  asm volatile("s_wait_asynccnt %0" :: "i"(N) : "memory");
#endif
}

// ---------------------------------------------------------------------------
// One block per (roi, py): 7 px stages, double-buffered async corner fetch.
// Each thread owns 4 contiguous channels; every wave consumes only the LDS
// bytes its own async ops wrote, so per-wave ASYNCcnt waits suffice (no
// barriers anywhere).
// ---------------------------------------------------------------------------
__global__ __launch_bounds__(THREADS)
void roi_pool_async_kernel(const float* __restrict__ img,
                           const int*   __restrict__ rois,
                           float*       __restrict__ out) {
  __shared__ float lds[2][4][CHAN];   // 2 stages x 4 corners x 1024 ch = 32 KB

  const int py  = blockIdx.x;         // 0..6
  const int r   = blockIdx.y;         // 0..511
  const int tid = threadIdx.x;
  const int c0  = tid * 4;            // this thread's 4 channels

  const int x = rois[r * 4 + 0];
  const int y = rois[r * 4 + 1];
  const int w = rois[r * 4 + 2];
  const int h = rois[r * 4 + 3];
  const float hf = (float)h;
  const float wf = (float)w;

  // y-direction sample (fixed for this block) — matches TF half-pixel centers
  float ysf = ((float)py + 0.5f) * hf / (float)POOL_P - 0.5f;
  ysf = fminf(fmaxf(ysf, 0.0f), hf - 1.0f);
  const int   y0  = (int)floorf(ysf);
  const float fy  = ysf - (float)y0;
  const int   y1  = min(y0 + 1, h - 1);
  const int   ay0 = y + y0;
  const int   ay1 = y + y1;

  // x-direction samples for all 7 px (kept in registers via full unroll)
  int   ax0[POOL_P], ax1[POOL_P];
  float fxv[POOL_P];
#pragma unroll
  for (int px = 0; px < POOL_P; ++px) {
    float xsf = ((float)px + 0.5f) * wf / (float)POOL_P - 0.5f;
    xsf = fminf(fmaxf(xsf, 0.0f), wf - 1.0f);
    const int x0 = (int)floorf(xsf);
    fxv[px] = xsf - (float)x0;
    const int x1 = min(x0 + 1, w - 1);
    ax0[px] = x + x0;
    ax1[px] = x + x1;
  }

  const size_t rowStride = (size_t)IMG_W * CHAN;
  const float* imgc = img + c0;       // per-thread channel base

  auto issue = [&](int px, int buf) {
    const size_t r0 = (size_t)ay0 * rowStride;
    const size_t r1 = (size_t)ay1 * rowStride;
    const size_t cA = (size_t)ax0[px] * CHAN;
    const size_t cB = (size_t)ax1[px] * CHAN;
    async_copy_b128(imgc + r0 + cA, &lds[buf][0][c0]);
    async_copy_b128(imgc + r0 + cB, &lds[buf][1][c0]);
    async_copy_b128(imgc + r1 + cA, &lds[buf][2][c0]);
    async_copy_b128(imgc + r1 + cB, &lds[buf][3][c0]);
  };

  issue(0, 0);  // prime the pipeline

  float* outp = out + ((size_t)r * POOL_P + py) * POOL_P * CHAN + c0;
  const float wy1 = fy, wy0 = 1.0f - fy;

#pragma unroll
  for (int px = 0; px < POOL_P; ++px) {
    if (px + 1 < POOL_P) {
      issue(px + 1, (px + 1) & 1);    // keep next stage in flight
      wait_asynccnt<4>();             // async ops retire in order: oldest 4 done
    } else {
      wait_asynccnt<0>();
    }

    const int b = px & 1;
    const float4 g00 = *(const float4*)&lds[b][0][c0];
    const float4 g01 = *(const float4*)&lds[b][1][c0];
    const float4 g10 = *(const float4*)&lds[b][2][c0];
    const float4 g11 = *(const float4*)&lds[b][3][c0];

    const float wx1 = fxv[px], wx0 = 1.0f - fxv[px];
    v4f_t res;
    res.x = (g00.x * wx0 + g01.x * wx1) * wy0 + (g10.x * wx0 + g11.x * wx1) * wy1;
    res.y = (g00.y * wx0 + g01.y * wx1) * wy0 + (g10.y * wx0 + g11.y * wx1) * wy1;
    res.z = (g00.z * wx0 + g01.z * wx1) * wy0 + (g10.z * wx0 + g11.z * wx1) * wy1;
    res.w = (g00.w * wx0 + g01.w * wx1) * wy0 + (g10.w * wx0 + g11.w * wx1) * wy1;

    // Streaming (non-temporal) store: output is write-once; keep the 64 MB
    // image resident in the 192 MB L2 instead of letting 100 MB of stores
    // churn it.
    __builtin_nontemporal_store(res, (v4f_t*)(outp + (size_t)px * CHAN));
  }
}

extern "C" void kernel_launch(void* const* d_in, const int* in_sizes, int n_in,
                              void* d_out, int out_size, void* d_ws, size_t ws_size,
                              hipStream_t stream) {
  (void)in_sizes; (void)n_in; (void)out_size; (void)d_ws; (void)ws_size;
  const float* img  = (const float*)d_in[0];   // (1,128,128,1024) f32
  const int*   rois = (const int*)d_in[1];     // (1,512,4) i32 (x,y,w,h)
  float*       out  = (float*)d_out;           // (1,512,7,7,1024) f32

  dim3 grid(POOL_P, NUM_ROIS, 1);              // (py, roi)
  roi_pool_async_kernel<<<grid, THREADS, 0, stream>>>(img, rois, out);
}